// DeEmph_44049184588381
// MI455X (gfx1250) — compile-verified
//
#include <hip/hip_runtime.h>

typedef float v2f __attribute__((ext_vector_type(2)));
typedef float v8f __attribute__((ext_vector_type(8)));

#define COEFF 0.95f
#define T_LEN 524288
#define N_ROWS 64
#define CHUNK 16384                      // output samples per block
#define HALO 4096                        // warmup samples (a^4096 == 0 in f32)
#define TOTAL (CHUNK + HALO)             // 20480
#define NSEG (TOTAL / 16)                // 1280 segments of 16 samples
#define NTILE (TOTAL / 256)              // 80 WMMA tiles (16 segments each)
#define THREADS 256
#define WAVES (THREADS / 32)             // 8 wave32s
#define TILES_PER_WAVE (NTILE / WAVES)   // 10 (interleaved: tile = wave + 8*tt)
#define HALO_TT 2                        // tt = 0,1 are halo tiles for every wave
#define OUT_TT (TILES_PER_WAVE - HALO_TT) // 8 live accumulators
#define CPT (NSEG / THREADS)             // 5 carries per thread
#define PAD_SIZE (TOTAL + NSEG)          // 21760 padded floats

// a^e for small runtime e (lane-dependent), e in [0,15]
__device__ __forceinline__ float apow_nz(int e) {
    if (e < 0) return 0.0f;
    float p = 1.0f;
    for (int k = 0; k < e; ++k) p *= COEFF;
    return p;
}

__global__ __launch_bounds__(THREADS)
void deemph_wmma_kernel(const float* __restrict__ x, float* __restrict__ y) {
    __shared__ float buf[PAD_SIZE];   // padded input, later reused for output
    __shared__ float cs[NSEG];        // raw segment carries
    __shared__ float ss[NSEG];        // inclusive-scanned carries
    __shared__ float tb[THREADS];     // per-thread partial carries

    const int tid  = threadIdx.x;
    const int lane = tid & 31;
    const int wave = tid >> 5;
    const int col  = lane & 15;       // matrix column / A-row (M) index
    const int hi   = lane >> 4;       // 0: lanes 0-15, 1: lanes 16-31

    const int row   = blockIdx.y;
    const int chunk = blockIdx.x;
    const long long rowBase    = (long long)row * T_LEN;
    const long long chunkStart = (long long)chunk * CHUNK;

    // ---- compile-time power table a^0..a^16 (constant-indexed only) ----
    float PA[17];
    PA[0] = 1.0f;
#pragma unroll
    for (int i = 1; i <= 16; ++i) PA[i] = PA[i - 1] * COEFF;
    const float r  = PA[16];          // a^16 : segment-to-segment ratio
    const float r5 = ((r * r) * (r * r)) * r;  // ratio across one thread's 5 carries

    // ---- Phase 0: global -> padded LDS (coalesced float4), halo zero-filled ----
#pragma unroll
    for (int it = 0; it < TOTAL / 4 / THREADS; ++it) {   // 20 iterations
        int t4 = (it * THREADS + tid) * 4;
        long long g = chunkStart - HALO + t4;            // <0 only for chunk 0
        float4 v;
        if (g >= 0) v = *(const float4*)(x + rowBase + g);
        else        v = make_float4(0.f, 0.f, 0.f, 0.f);
        int p0 = t4 + (t4 >> 4);                         // pad: stride 16 -> 17
        buf[p0 + 0] = v.x; buf[p0 + 1] = v.y; buf[p0 + 2] = v.z; buf[p0 + 3] = v.w;
    }

    // ---- A matrix: L[:, 4kk + {0,1,2,3}] per ISA 16x4 f32 layout ----
    // VGPR0: lanes0-15 K=j0, lanes16-31 K=j0+2 ; VGPR1: K=j0+1 / j0+3
    v2f amat[4];
#pragma unroll
    for (int kk = 0; kk < 4; ++kk) {
        int j0 = 4 * kk + (hi ? 2 : 0);
        amat[kk][0] = apow_nz(col - j0);
        amat[kk][1] = apow_nz(col - (j0 + 1));
    }
    __syncthreads();

    // ---- Phase A: Z = L @ X per tile via 4 chained v_wmma_f32_16x16x4_f32 ----
    // Interleaved tile assignment: tile = wave + 8*tt; tt<2 are halo tiles whose
    // accumulators die right after the carry extract (keeps only 8 live v8f).
    v8f acc[OUT_TT];
#pragma unroll
    for (int tt = 0; tt < TILES_PER_WAVE; ++tt) {
        int tile = wave + WAVES * tt;
        int base = tile * 272 + col * 17;                // padded addr of (tile, seg=col, k=0)
        v8f d = {0.f, 0.f, 0.f, 0.f, 0.f, 0.f, 0.f, 0.f};
#pragma unroll
        for (int kk = 0; kk < 4; ++kk) {
            int k0 = 4 * kk + (hi ? 2 : 0);              // B 4x16 layout mirrors A
            v2f b;
            b[0] = buf[base + k0];
            b[1] = buf[base + k0 + 1];
            d = __builtin_amdgcn_wmma_f32_16x16x4_f32(
                    /*neg_a=*/false, amat[kk], /*neg_b=*/false, b,
                    /*c_mod=*/(short)0, d, /*reuse_a=*/false, /*reuse_b=*/false);
        }
        // carry = Z[15][seg] lives in VGPR7 of lanes 16-31 (M = 7 + 8)
        if (hi) cs[tile * 16 + col] = d[7];
        if (tt >= HALO_TT) acc[tt - HALO_TT] = d;        // compile-time after unroll
    }
    __syncthreads();

    // ---- Phase C: affine scan of 1280 segment carries, ratio r = a^16 ----
    {
        int s0 = tid * CPT;                              // 256 threads * 5 = 1280
        float v0 = cs[s0], v1 = cs[s0 + 1], v2 = cs[s0 + 2],
              v3 = cs[s0 + 3], v4 = cs[s0 + 4];
        // local inclusive compose (factor r^5)
        tb[tid] = fmaf(r, fmaf(r, fmaf(r, fmaf(r, v0, v1), v2), v3), v4);
        __syncthreads();
        if (tid == 0) {
            float P = 0.f;
            for (int u = 0; u < THREADS; ++u) {
                float b = tb[u];
                tb[u] = P;                               // exclusive prefix
                P = fmaf(r5, P, b);
            }
        }
        __syncthreads();
        float Pt = tb[tid];
        float S0 = fmaf(r, Pt, v0);
        float S1 = fmaf(r, S0, v1);
        float S2 = fmaf(r, S1, v2);
        float S3 = fmaf(r, S2, v3);
        float S4 = fmaf(r, S3, v4);
        ss[s0] = S0; ss[s0 + 1] = S1; ss[s0 + 2] = S2;
        ss[s0 + 3] = S3; ss[s0 + 4] = S4;                // inclusive carry scan
    }
    __syncthreads();

    // ---- Phase D: fixup y = z + a^(i+1)*Cin, transpose-store to padded LDS ----
#pragma unroll
    for (int tt = HALO_TT; tt < TILES_PER_WAVE; ++tt) {
        int tile = wave + WAVES * tt;                    // always >= 16: non-halo
        int seg = tile * 16 + col;
        float Cin = ss[seg - 1];                         // seg >= 256, no guard needed
        v8f d = acc[tt - HALO_TT];
        int base = tile * 272 + col * 17 + (hi ? 8 : 0); // M = v + 8*hi
#pragma unroll
        for (int v = 0; v < 8; ++v) {
            float pm = hi ? PA[v + 9] : PA[v + 1];       // a^(M+1), constants
            buf[base + v] = fmaf(pm, Cin, d[v]);
        }
    }
    __syncthreads();

    // ---- Phase E: coalesced float4 stores of the non-halo region ----
#pragma unroll
    for (int it = 0; it < CHUNK / 4 / THREADS; ++it) {   // 16 iterations
        int t4 = HALO + (it * THREADS + tid) * 4;
        int p0 = t4 + (t4 >> 4);
        float4 v = make_float4(buf[p0], buf[p0 + 1], buf[p0 + 2], buf[p0 + 3]);
        *(float4*)(y + rowBase + chunkStart + (t4 - HALO)) = v;
    }
}

extern "C" void kernel_launch(void* const* d_in, const int* in_sizes, int n_in,
                              void* d_out, int out_size, void* d_ws, size_t ws_size,
                              hipStream_t stream) {
    (void)in_sizes; (void)n_in; (void)out_size; (void)d_ws; (void)ws_size;
    const float* x = (const float*)d_in[0];
    float* y = (float*)d_out;
    dim3 grid(T_LEN / CHUNK, N_ROWS);   // (32 chunks, 64 rows) = 2048 blocks
    deemph_wmma_kernel<<<grid, dim3(THREADS), 0, stream>>>(x, y);
}